// TP_nonlinearity_23837068492872
// MI455X (gfx1250) — compile-verified
//
#include <hip/hip_runtime.h>

typedef float v2f __attribute__((ext_vector_type(2)));
typedef float v8f __attribute__((ext_vector_type(8)));

#define NTRI  42
#define BATCH 256
#define MULC  32
#define LMAX  4
#define KPAD  12           // max 2*l1+1 = 9, padded to 12 (3 WMMA K-steps of 4)
#define MAXT  10           // max triples per l3 (l3 == 2 has 10)

struct TripleDesc { const float* w3j; int l1; int l2; };

struct ParamsL {
  const float* x[5];  // x[l]: [B, MUL, 2l+1]
  float* out;
  unsigned long long base;   // flat element offset of this l3's output segment
  TripleDesc tr[MAXT];
};

// One block = one (batch, triple-of-this-l3). 128 threads = 4 waves.
// Stage 1: zero-padded A ([32][KPAD]) and Y ([KPAD][N]) in LDS, K-pair interleaved
//          so stage 2 reads ds_load_b64.  Y[m, j*D3+M] = sum_n w3j[m,n,M]*x2[b,j,n].
// Stage 2: out[i,c] = sum_m A[i,m]*Y[m,c] -> 3 chained v_wmma_f32_16x16x4_f32 per
//          16x16 tile; D3/N/NT compile-time -> immediate-offset DS loads & stores,
//          scalar (readfirstlane) tile loop keeps EXEC all-ones around WMMA.
template <int D3, int NT>
__launch_bounds__(128)
__global__ void tp_wmma_kernel(ParamsL P) {
  constexpr int N = MULC * D3;                     // 32..288
  constexpr int BSTRIDE = NT * MULC * MULC * D3;   // per-batch elements for this l3
  const int blk = blockIdx.x;
  const int b = blk / NT;
  const int t = blk - b * NT;
  const TripleDesc td = P.tr[t];

  const int K1 = 2 * td.l1 + 1;
  const int K2 = 2 * td.l2 + 1;
  const int tid = threadIdx.x;

  __shared__ float Ys[(KPAD / 2) * N * 2];
  __shared__ float As[MULC * KPAD];

  const float* __restrict__ x1 = P.x[td.l1] + (size_t)b * MULC * K1;
  const float* __restrict__ x2 = P.x[td.l2] + (size_t)b * MULC * K2;
  const float* __restrict__ w  = td.w3j;

  // ---- Stage 1a: zero-padded A into LDS ----
  for (int idx = tid; idx < MULC * KPAD; idx += 128) {
    const int row = idx / KPAD;
    const int k   = idx - row * KPAD;
    As[(row * (KPAD / 2) + (k >> 1)) * 2 + (k & 1)] = (k < K1) ? x1[row * K1 + k] : 0.f;
  }

  // ---- Stage 1b: Y into LDS (rows m >= K1 zeroed) ----
  for (int idx = tid; idx < KPAD * N; idx += 128) {
    const int m = idx / N;
    const int c = idx - m * N;
    float acc = 0.f;
    if (m < K1) {
      const int j = c / D3;
      const int M = c - j * D3;
      const float* wr = w + (m * K2) * D3 + M;
      const float* xr = x2 + j * K2;
      for (int n = 0; n < K2; ++n) acc += wr[n * D3] * xr[n];
    }
    Ys[((m >> 1) * N + c) * 2 + (m & 1)] = acc;
  }
  __syncthreads();

  // ---- Stage 2: WMMA tiles ----
  const int lane = tid & 31;
  const int half = lane >> 4;           // 0: K ks+0/1, 1: K ks+2/3
  const int l16  = lane & 15;
  const int wv   = __builtin_amdgcn_readfirstlane(tid >> 5);  // scalar wave id
  const int mtile = wv >> 1;            // waves 0,1 -> rows 0-15; 2,3 -> rows 16-31
  const int nt0   = wv & 1;             // N-tile parity for this wave
  const int row   = mtile * 16 + l16;   // A row (matrix M index)
  const int rbase = mtile * 16 + half * 8;

  const float* Arow = &As[row * KPAD];
  const v2f a0 = *(const v2f*)&Arow[(0 + half) * 2];
  const v2f a1 = *(const v2f*)&Arow[(2 + half) * 2];
  const v2f a2 = *(const v2f*)&Arow[(4 + half) * 2];

  // per-lane bases; all loop offsets below are compile-time immediates
  const float* Yb = &Ys[(half * N + nt0 * 16 + l16) * 2];
  float* __restrict__ outp = P.out + (size_t)P.base + (size_t)t * (MULC * MULC * D3)
                           + (size_t)b * BSTRIDE
                           + (size_t)rbase * N + nt0 * 16 + l16;

  #pragma unroll
  for (int i = 0; i < D3; ++i) {        // this wave's N-tiles: nt = nt0 + 2*i
    const v2f b0 = *(const v2f*)&Yb[(0 * N) * 2 + 64 * i];
    const v2f b1 = *(const v2f*)&Yb[(2 * N) * 2 + 64 * i];
    const v2f b2 = *(const v2f*)&Yb[(4 * N) * 2 + 64 * i];

    v8f acc = {0.f, 0.f, 0.f, 0.f, 0.f, 0.f, 0.f, 0.f};
    acc = __builtin_amdgcn_wmma_f32_16x16x4_f32(false, a0, false, b0, (short)0, acc, false, false);
    acc = __builtin_amdgcn_wmma_f32_16x16x4_f32(false, a1, false, b1, (short)0, acc, false, false);
    acc = __builtin_amdgcn_wmma_f32_16x16x4_f32(false, a2, false, b2, (short)0, acc, false, false);

    #pragma unroll
    for (int v = 0; v < 8; ++v)
      outp[v * N + 32 * i] = acc[v];    // D: VGPR v -> row rbase+v, col nt*16+l16
  }
}

extern "C" void kernel_launch(void* const* d_in, const int* in_sizes, int n_in,
                              void* d_out, int out_size, void* d_ws, size_t ws_size,
                              hipStream_t stream) {
  (void)in_sizes; (void)out_size; (void)d_ws; (void)ws_size;

  // Enumerate triples exactly like the reference: l1<=l2, |l1-l2| <= l3 <= min(l1+l2, 4)
  int l1s[NTRI], l2s[NTRI], l3s[NTRI];
  int nt = 0;
  for (int l1 = 0; l1 <= LMAX; ++l1)
    for (int l2 = l1; l2 <= LMAX; ++l2) {
      const int hi = (l1 + l2 < LMAX) ? (l1 + l2) : LMAX;
      for (int l3 = l2 - l1; l3 <= hi; ++l3) {
        l1s[nt] = l1; l2s[nt] = l2; l3s[nt] = l3; ++nt;
      }
    }

  // Per-l3 channel counts and flat output bases (tuple concatenated in l3 order)
  long long ccount[5] = {0, 0, 0, 0, 0};
  for (int i = 0; i < nt; ++i) ccount[l3s[i]] += MULC * MULC;
  long long base[5]; long long run = 0;
  for (int l3 = 0; l3 <= LMAX; ++l3) {
    base[l3] = run;
    run += (long long)BATCH * ccount[l3] * (2 * l3 + 1);
  }

  // Build per-l3 parameter blocks; within-l3 order == enumeration order
  ParamsL P[5];
  int cnt[5] = {0, 0, 0, 0, 0};
  const bool separate_w3j = (n_in >= 5 + NTRI);
  long long w3joff = 0;
  for (int l3 = 0; l3 <= LMAX; ++l3) {
    for (int l = 0; l < 5; ++l) P[l3].x[l] = (const float*)d_in[l];
    P[l3].out  = (float*)d_out;
    P[l3].base = (unsigned long long)base[l3];
  }
  for (int i = 0; i < nt; ++i) {
    const int l1 = l1s[i], l2 = l2s[i], l3 = l3s[i];
    TripleDesc& td = P[l3].tr[cnt[l3]++];
    td.l1 = l1; td.l2 = l2;
    if (separate_w3j)
      td.w3j = (const float*)d_in[5 + i];                 // one leaf buffer per triple
    else
      td.w3j = (const float*)d_in[5] + w3joff;            // single concatenated buffer
    w3joff += (long long)(2 * l1 + 1) * (2 * l2 + 1) * (2 * l3 + 1);
  }

  // Triples per l3: {5, 8, 10, 10, 9}
  hipLaunchKernelGGL((tp_wmma_kernel<1, 5>),  dim3(BATCH * 5),  dim3(128), 0, stream, P[0]);
  hipLaunchKernelGGL((tp_wmma_kernel<3, 8>),  dim3(BATCH * 8),  dim3(128), 0, stream, P[1]);
  hipLaunchKernelGGL((tp_wmma_kernel<5, 10>), dim3(BATCH * 10), dim3(128), 0, stream, P[2]);
  hipLaunchKernelGGL((tp_wmma_kernel<7, 9>),  dim3(BATCH * 9),  dim3(128), 0, stream, P[3]);
  hipLaunchKernelGGL((tp_wmma_kernel<9, 9>),  dim3(BATCH * 9),  dim3(128), 0, stream, P[4]);
}